// RhoAdjustedGAT_44667659879044
// MI455X (gfx1250) — compile-verified
//
#include <hip/hip_runtime.h>
#include <hip/hip_bf16.h>

// ---------------- problem constants (match reference) ----------------
#define NNODES   50000
#define NEDGES   800000
#define EPRIME   (NEDGES + NNODES)   // self-loops appended
#define IN_DIM   128
#define HID      64
#define HEADS    4
#define OUT_DIM  2
#define NGRAPHS  500
#define NEG_SLOPE 0.2f

typedef float v2f __attribute__((ext_vector_type(2)));
typedef float v8f __attribute__((ext_vector_type(8)));

static __device__ __forceinline__ unsigned f2key(float f) {
  unsigned b = __float_as_uint(f);
  return (b & 0x80000000u) ? ~b : (b | 0x80000000u);   // monotone as unsigned
}
static __device__ __forceinline__ float key2f(unsigned u) {
  return (u & 0x80000000u) ? __uint_as_float(u & 0x7FFFFFFFu) : __uint_as_float(~u);
}

// ---------------- zero fill ----------------
__global__ void zero_u32_kernel(unsigned* __restrict__ p, long long n) {
  long long i = (long long)blockIdx.x * blockDim.x + threadIdx.x;
  if (i < n) p[i] = 0u;
}

// ---------------- fp32 WMMA GEMM with LDS-staged B panel -------------------------
// C[rows x M] = A[rows x K] @ B[K x M] + bias.  Block = 256 threads = 8 waves.
// Each block owns 8 row-tiles x one 64-col group. The K x 64 B panel is staged in
// LDS, k-pair interleaved (pair p = k>>1) with row stride 130 dwords (bank pad),
// so a WMMA B fragment {B[k][c], B[k+1][c]} is one aligned ds_load_b64.
// A fragments: clause of 4 global_load_b64 per 16-k chunk, consumed by 16 WMMAs.
#define LDSB_STRIDE 130

template <int K>
__global__ void gemm_wmma_lds_kernel(const float* __restrict__ A,
                                     const float* __restrict__ B,
                                     const float* __restrict__ bias,
                                     float* __restrict__ C,
                                     int M, int rowTiles, int colGroups) {
  __shared__ float ldsB[(K / 2) * LDSB_STRIDE];

  int colGroup = blockIdx.x % colGroups;
  int rowBlock = blockIdx.x / colGroups;
  int colBase  = colGroup * 64;

  // ---- cooperative stage of B panel: K x 64, k-pair interleaved ----
  for (int i = threadIdx.x; i < K * 64; i += 256) {
    int k = i >> 6, c = i & 63;
    ldsB[(k >> 1) * LDSB_STRIDE + c * 2 + (k & 1)] = B[(size_t)k * M + colBase + c];
  }
  __syncthreads();

  int waveInBlk = threadIdx.x >> 5;
  int lane      = threadIdx.x & 31;
  int rowTile   = rowBlock * 8 + waveInBlk;
  if (rowTile >= rowTiles) return;                    // wave-uniform: EXEC all-1s below

  int r  = lane & 15;                                 // row-in-tile (A) / col-in-tile (B,C)
  int hi = lane >> 4;                                 // lane-half: K pair select / M half
  const float* Arow = A + (size_t)(rowTile * 16 + r) * K;

  v8f acc0 = {}, acc1 = {}, acc2 = {}, acc3 = {};
#pragma unroll 1
  for (int k0 = 0; k0 < K; k0 += 16) {
    __builtin_prefetch(Arow + k0 + 16, 0, 0);         // global_prefetch_b8 (next chunk)
    // clause of 4 A-fragment loads for this 16-k chunk
    v2f a0 = *(const v2f*)(Arow + k0 + 0  + 2 * hi);
    v2f a1 = *(const v2f*)(Arow + k0 + 4  + 2 * hi);
    v2f a2 = *(const v2f*)(Arow + k0 + 8  + 2 * hi);
    v2f a3 = *(const v2f*)(Arow + k0 + 12 + 2 * hi);
    v2f af[4] = {a0, a1, a2, a3};
#pragma unroll
    for (int sub = 0; sub < 4; ++sub) {
      int p = (k0 >> 1) + 2 * sub + hi;               // ISA A/B layout: lanes16-31 = K+2,K+3
      const float* bb = &ldsB[p * LDSB_STRIDE + r * 2];
      v2f a = af[sub];
      acc0 = __builtin_amdgcn_wmma_f32_16x16x4_f32(false, a, false, *(const v2f*)(bb + 0),
                                                   (short)0, acc0, false, false);
      acc1 = __builtin_amdgcn_wmma_f32_16x16x4_f32(false, a, false, *(const v2f*)(bb + 32),
                                                   (short)0, acc1, false, false);
      acc2 = __builtin_amdgcn_wmma_f32_16x16x4_f32(false, a, false, *(const v2f*)(bb + 64),
                                                   (short)0, acc2, false, false);
      acc3 = __builtin_amdgcn_wmma_f32_16x16x4_f32(false, a, false, *(const v2f*)(bb + 96),
                                                   (short)0, acc3, false, false);
    }
  }

  // C/D layout: VGPR v -> row v + 8*hi, col = lane&15
  const float* bp = bias + colBase + r;
#pragma unroll
  for (int v = 0; v < 8; ++v) {
    float* Crow = C + (size_t)(rowTile * 16 + v + 8 * hi) * M + colBase + r;
    Crow[0]  = acc0[v] + bp[0];
    Crow[16] = acc1[v] + bp[16];
    Crow[32] = acc2[v] + bp[32];
    Crow[48] = acc3[v] + bp[48];
  }
}

// ---------------- per-node attention logits s = h.a_src, d = h.a_dst ----------------
__global__ void node_sd_kernel(const float* __restrict__ hfeat,
                               const float* __restrict__ asrc,
                               const float* __restrict__ adst,
                               float* __restrict__ s, float* __restrict__ d,
                               int H, int D) {
  long long idx = (long long)blockIdx.x * blockDim.x + threadIdx.x;
  if (idx >= (long long)NNODES * H) return;
  int n = (int)(idx / H), h = (int)(idx % H);
  const float* hp = hfeat + (size_t)n * H * D + (size_t)h * D;
  const float* as = asrc + (size_t)h * D;
  const float* ad = adst + (size_t)h * D;
  float ss = 0.f, dd = 0.f;
  for (int i = 0; i < D; ++i) { float v = hp[i]; ss += v * as[i]; dd += v * ad[i]; }
  s[idx] = ss; d[idx] = dd;
}

// ---------------- pass A: segment max of leaky_relu(s[src]+d[dst]) over dst ----------
__global__ void edge_max_kernel(const int* __restrict__ ei,
                                const float* __restrict__ s,
                                const float* __restrict__ dv,
                                unsigned* __restrict__ mbits, int H) {
  long long t = (long long)blockIdx.x * blockDim.x + threadIdx.x;
  if (t >= (long long)EPRIME * H) return;
  int e = (int)(t / H), h = (int)(t % H);
  int src, dst;
  if (e < NEDGES) { src = ei[e]; dst = ei[NEDGES + e]; } else { src = dst = e - NEDGES; }
  float v = s[(size_t)src * H + h] + dv[(size_t)dst * H + h];
  v = (v > 0.f) ? v : NEG_SLOPE * v;
  atomicMax(&mbits[(size_t)dst * H + h], f2key(v));
}

// ---------------- pass B: accumulate den += w, acc += w * h[src] ----------------
__global__ void edge_accum_kernel(const int* __restrict__ ei,
                                  const float* __restrict__ s,
                                  const float* __restrict__ dv,
                                  const unsigned* __restrict__ mbits,
                                  const float* __restrict__ hfeat,
                                  float* __restrict__ den,
                                  float* __restrict__ accum,
                                  int H, int D) {
  long long t = (long long)blockIdx.x * blockDim.x + threadIdx.x;
  if (t >= (long long)EPRIME * H) return;
  int e = (int)(t / H), h = (int)(t % H);
  int src, dst;
  if (e < NEDGES) { src = ei[e]; dst = ei[NEDGES + e]; } else { src = dst = e - NEDGES; }
  float v = s[(size_t)src * H + h] + dv[(size_t)dst * H + h];
  v = (v > 0.f) ? v : NEG_SLOPE * v;
  float m = key2f(mbits[(size_t)dst * H + h]);
  float w = __expf(v - m);
  atomicAdd(&den[(size_t)dst * H + h], w);
  const float* hs = hfeat + (size_t)src * H * D + (size_t)h * D;
  float* ad = accum + (size_t)dst * H * D + (size_t)h * D;
  for (int i = 0; i < D; ++i) atomicAdd(&ad[i], w * hs[i]);
}

// ---------------- pass C: out = elu(acc/den + bias) (element-wise, in-place safe) ----
__global__ void norm_elu_kernel(const float* __restrict__ accum,
                                const float* __restrict__ den,
                                const float* __restrict__ bias,
                                float* __restrict__ out,
                                int H, int D, long long total) {
  long long idx = (long long)blockIdx.x * blockDim.x + threadIdx.x;
  if (idx >= total) return;
  int HD = H * D;
  int n = (int)(idx / HD), c = (int)(idx % HD), h = c / D;
  float v = accum[idx] / den[(size_t)n * H + h] + bias[c];
  out[idx] = (v > 0.f) ? v : (__expf(v) - 1.0f);
}

// ---------------- global mean pool (sum + count) ----------------
__global__ void pool_kernel(const float* __restrict__ h2,
                            const int* __restrict__ batch,
                            float* __restrict__ pool, float* __restrict__ cnt) {
  long long idx = (long long)blockIdx.x * blockDim.x + threadIdx.x;
  if (idx >= (long long)NNODES * HID) return;
  int n = (int)(idx >> 6), dd = (int)(idx & 63);
  int g = batch[n];
  atomicAdd(&pool[(size_t)g * HID + dd], h2[idx]);
  if (dd == 0) atomicAdd(&cnt[g], 1.0f);
}

// ---------------- final FC: out[g,o] = mean_pool[g] . fcW[:,o] + fcb[o] ----------------
__global__ void fc_kernel(const float* __restrict__ pool, const float* __restrict__ cnt,
                          const float* __restrict__ W, const float* __restrict__ b,
                          float* __restrict__ out) {
  int idx = blockIdx.x * blockDim.x + threadIdx.x;
  if (idx >= NGRAPHS * OUT_DIM) return;
  int g = idx >> 1, o = idx & 1;
  float c = cnt[g]; c = (c > 1.0f) ? c : 1.0f;
  float sum = 0.f;
  for (int k = 0; k < HID; ++k) sum += (pool[(size_t)g * HID + k] / c) * W[k * OUT_DIM + o];
  out[idx] = sum + b[o];
}

// ---------------- host-side orchestration ----------------
static inline int cdiv_ll(long long a, int b) { return (int)((a + b - 1) / b); }

extern "C" void kernel_launch(void* const* d_in, const int* in_sizes, int n_in,
                              void* d_out, int out_size, void* d_ws, size_t ws_size,
                              hipStream_t stream) {
  const float* x      = (const float*)d_in[0];
  const int*   eidx   = (const int*)d_in[1];
  const int*   batch  = (const int*)d_in[2];
  const float* W1     = (const float*)d_in[3];
  const float* a_src1 = (const float*)d_in[4];
  const float* a_dst1 = (const float*)d_in[5];
  const float* b1     = (const float*)d_in[6];
  const float* W2     = (const float*)d_in[7];
  const float* a_src2 = (const float*)d_in[8];
  const float* a_dst2 = (const float*)d_in[9];
  const float* b2     = (const float*)d_in[10];
  const float* fcW    = (const float*)d_in[11];
  const float* fcb    = (const float*)d_in[12];
  float* out = (float*)d_out;

  // workspace partition (floats)
  float* ws = (float*)d_ws;
  size_t off = 0;
  float*    hpre = ws + off; off += (size_t)NNODES * (HEADS * HID);  // layer1 W-feats; layer2 reuses [0, N*64)
  float*    acc  = ws + off; off += (size_t)NNODES * (HEADS * HID);  // agg acc; becomes x2 in-place; reused layer2
  float*    s1   = ws + off; off += (size_t)NNODES * HEADS;
  float*    d1   = ws + off; off += (size_t)NNODES * HEADS;
  float*    den1 = ws + off; off += (size_t)NNODES * HEADS;
  unsigned* m1   = (unsigned*)(ws + off); off += (size_t)NNODES * HEADS;
  float*    s2   = ws + off; off += (size_t)NNODES;
  float*    d2   = ws + off; off += (size_t)NNODES;
  float*    den2 = ws + off; off += (size_t)NNODES;
  unsigned* m2   = (unsigned*)(ws + off); off += (size_t)NNODES;
  float*    pool = ws + off; off += (size_t)NGRAPHS * HID;
  float*    cnt  = ws + off; off += (size_t)NGRAPHS;

  const int BLK = 256;
  const int rowTiles  = NNODES / 16;                 // 3125
  const int rowBlocks = (rowTiles + 7) / 8;          // 391

  // ================= Layer 1 (H=4, D=64, K=128, M=256) =================
  {
    int colGroups = (HEADS * HID) / 64;              // 4
    gemm_wmma_lds_kernel<IN_DIM><<<rowBlocks * colGroups, BLK, 0, stream>>>(
        x, W1, b1, hpre, HEADS * HID, rowTiles, colGroups);

    long long nh = (long long)NNODES * HEADS;
    node_sd_kernel<<<cdiv_ll(nh, BLK), BLK, 0, stream>>>(hpre, a_src1, a_dst1, s1, d1, HEADS, HID);

    zero_u32_kernel<<<cdiv_ll(nh, BLK), BLK, 0, stream>>>(m1, nh);
    zero_u32_kernel<<<cdiv_ll(nh, BLK), BLK, 0, stream>>>((unsigned*)den1, nh);
    long long nhd = (long long)NNODES * HEADS * HID;
    zero_u32_kernel<<<cdiv_ll(nhd, BLK), BLK, 0, stream>>>((unsigned*)acc, nhd);

    long long eh = (long long)EPRIME * HEADS;
    edge_max_kernel<<<cdiv_ll(eh, BLK), BLK, 0, stream>>>(eidx, s1, d1, m1, HEADS);
    edge_accum_kernel<<<cdiv_ll(eh, BLK), BLK, 0, stream>>>(eidx, s1, d1, m1, hpre, den1, acc, HEADS, HID);
    norm_elu_kernel<<<cdiv_ll(nhd, BLK), BLK, 0, stream>>>(acc, den1, b1, acc, HEADS, HID, nhd);
    // acc now holds x2 = elu(GAT1 output), [N, 256]
  }

  // ================= Layer 2 (H=1, D=64, K=256, M=64) =================
  {
    gemm_wmma_lds_kernel<HEADS * HID><<<rowBlocks, BLK, 0, stream>>>(
        acc, W2, b2, hpre, HID, rowTiles, 1);        // hpre[0, N*64) = h2 pre-feats

    long long nh = (long long)NNODES;
    node_sd_kernel<<<cdiv_ll(nh, BLK), BLK, 0, stream>>>(hpre, a_src2, a_dst2, s2, d2, 1, HID);

    zero_u32_kernel<<<cdiv_ll(nh, BLK), BLK, 0, stream>>>(m2, nh);
    zero_u32_kernel<<<cdiv_ll(nh, BLK), BLK, 0, stream>>>((unsigned*)den2, nh);
    long long nd = (long long)NNODES * HID;
    zero_u32_kernel<<<cdiv_ll(nd, BLK), BLK, 0, stream>>>((unsigned*)acc, nd);  // x2 dead after GEMM2

    long long eh = (long long)EPRIME;
    edge_max_kernel<<<cdiv_ll(eh, BLK), BLK, 0, stream>>>(eidx, s2, d2, m2, 1);
    edge_accum_kernel<<<cdiv_ll(eh, BLK), BLK, 0, stream>>>(eidx, s2, d2, m2, hpre, den2, acc, 1, HID);
    norm_elu_kernel<<<cdiv_ll(nd, BLK), BLK, 0, stream>>>(acc, den2, b2, acc, 1, HID, nd);
    // acc[0, N*64) now holds final node embeddings h2
  }

  // ================= mean pool + FC =================
  {
    long long pz = (long long)NGRAPHS * HID + NGRAPHS;
    zero_u32_kernel<<<cdiv_ll(pz, BLK), BLK, 0, stream>>>((unsigned*)pool, pz);
    long long nd = (long long)NNODES * HID;
    pool_kernel<<<cdiv_ll(nd, BLK), BLK, 0, stream>>>(acc, batch, pool, cnt);
    fc_kernel<<<cdiv_ll(NGRAPHS * OUT_DIM, BLK), BLK, 0, stream>>>(pool, cnt, fcW, fcb, out);
  }
}